// BoundaryWeightedLoss_67499706024381
// MI455X (gfx1250) — compile-verified
//
#include <hip/hip_runtime.h>
#include <stdint.h>

#define HH 256
#define WW 256
#define NB 4
#define SLICE_ELEMS (HH*WW)
#define NSLICE 8                 // B*C distinct EDT slices
#define NTOT (NB*2*HH*WW)        // 524288 channel-elements

typedef __attribute__((ext_vector_type(2))) float v2f;
typedef __attribute__((ext_vector_type(8))) float v8f;

__device__ __forceinline__ float log_sigmoid_f(float x) {
    // stable: min(x,0) - log1p(exp(-|x|))
    return fminf(x, 0.0f) - log1pf(expf(-fabsf(x)));
}

// ---------------- init: zero the 8 per-slice max slots -----------------
__global__ void init_kernel(unsigned* __restrict__ meta) {
    if (threadIdx.x < NSLICE) meta[threadIdx.x] = 0u;
}

// ------------- phase 1: per-column 1D distance, squared ----------------
// slice s = b*2+c; zeros = TP pixels of channel c (== TN of channel 1-c)
__global__ void edt_cols_kernel(const float* __restrict__ pred,
                                const float* __restrict__ label,
                                float* __restrict__ g2) {
    const int s = blockIdx.x;              // 0..7
    const int b = s >> 1, c = s & 1;
    const int w = threadIdx.x;             // column
    const float* p0 = pred  + (b*2+0)*SLICE_ELEMS + w;
    const float* p1 = pred  + (b*2+1)*SLICE_ELEMS + w;
    const float* l0 = label + (b*2+0)*SLICE_ELEMS + w;
    const float* l1 = label + (b*2+1)*SLICE_ELEMS + w;
    float* grow = g2 + s*SLICE_ELEMS + w;

    unsigned zb[8];
    int last = -513;                       // reference BIG = H+W+1 = 513
#pragma unroll
    for (int wd = 0; wd < 8; ++wd) {
        unsigned bits = 0u;
        for (int bit = 0; bit < 32; ++bit) {
            const int i = wd*32 + bit;
            const float a0 = p0[i*WW], a1 = p1[i*WW];
            const float b0 = l0[i*WW], b1 = l1[i*WW];
            const bool m0 = (a0 >= a1);    // argmax(sigmoid(pred)) == argmax(pred)
            const bool o0 = (b0 >= b1);    // jnp.argmax first-max tie rule
            const bool z  = (c == 0) ? (m0 && o0) : ((!m0) && (!o0));
            if (z) { last = i; bits |= (1u << bit); }
            grow[i*WW] = (float)(i - last);      // down-distance (<=768, exact fp32)
        }
        zb[wd] = bits;
    }
    int nxt = 513;
#pragma unroll
    for (int wd = 7; wd >= 0; --wd) {
        const unsigned bits = zb[wd];
        for (int bit = 31; bit >= 0; --bit) {
            const int i = wd*32 + bit;
            if ((bits >> bit) & 1u) nxt = i;
            const float gd = grow[i*WW];
            const float g  = fminf(gd, (float)(nxt - i));
            grow[i*WW] = g * g;                  // g^2 exact (< 2^24)
        }
    }
}

// ------------- phase 2: exact min-plus along W, per row ----------------
// One block per (slice,row). g^2 row staged into LDS via CDNA5 async load;
// inner loop reads LDS as b128 broadcasts (4 k per ds op).
__global__ void edt_rows_kernel(float* __restrict__ g2,
                                unsigned* __restrict__ meta) {
    __shared__ __align__(16) float sg2[WW];
    __shared__ float sred[256];
    const int s = blockIdx.x >> 8;
    const int i = blockIdx.x & 255;
    const int j = threadIdx.x;
    float* row = g2 + s*SLICE_ELEMS + i*WW;

    // async global->LDS copy of the 1KB row (tracked by ASYNCcnt)
    {
        unsigned lds = (unsigned)(unsigned long long)(&sg2[j]);
        unsigned long long src = (unsigned long long)(row + j);
        asm volatile("global_load_async_to_lds_b32 %0, %1, off"
                     :: "v"(lds), "v"(src) : "memory");
        asm volatile("s_wait_asynccnt 0" ::: "memory");
    }
    __syncthreads();

    const float fj = (float)j;
    float m = 3.402823466e+38f;
    const float4* sg4 = (const float4*)sg2;
#pragma unroll 4
    for (int k4 = 0; k4 < WW/4; ++k4) {
        const float4 g = sg4[k4];                 // ds_load_b128 broadcast
        const float kb = (float)(4*k4);
        const float d0 = fj - kb;
        const float d1 = fj - (kb + 1.0f);
        const float d2 = fj - (kb + 2.0f);
        const float d3 = fj - (kb + 3.0f);
        m = fminf(m, __builtin_fmaf(d0, d0, g.x)); // g^2 + (j-k)^2, exact ints
        m = fminf(m, __builtin_fmaf(d1, d1, g.y));
        m = fminf(m, __builtin_fmaf(d2, d2, g.z));
        m = fminf(m, __builtin_fmaf(d3, d3, g.w));
    }
    row[j] = m;                                    // D^2 in place (row is in LDS)

    // block max of D^2 -> per-slice max (uint compare valid for +floats)
    sred[j] = m;
    __syncthreads();
    for (int off = 128; off > 0; off >>= 1) {
        if (j < off) sred[j] = fmaxf(sred[j], sred[j + off]);
        __syncthreads();
    }
    if (j == 0) atomicMax(&meta[s], __float_as_uint(sred[0]));
}

// ------------- fused alpha*CE + WMMA wave reduction --------------------
__global__ void loss_kernel(const float* __restrict__ pred,
                            const float* __restrict__ label,
                            const float* __restrict__ d2,
                            const unsigned* __restrict__ meta,
                            float* __restrict__ psum,
                            unsigned* __restrict__ pcnt) {
    __shared__ float swave[8];
    __shared__ unsigned scnt[256];
    const int t = blockIdx.x * 256 + threadIdx.x;     // 0..262143
    float vals[2];
    unsigned cnt = 0;
#pragma unroll
    for (int r = 0; r < 2; ++r) {
        const int e = t + r * (NTOT/2);
        const int b = e >> 17;
        const int c = (e >> 16) & 1;
        const int h = (e >> 8) & 255;
        const int w = e & 255;
        const int base = b*2*SLICE_ELEMS + h*WW + w;
        const float a0 = pred[base],  a1 = pred[base + SLICE_ELEMS];
        const float b0 = label[base], b1 = label[base + SLICE_ELEMS];
        const bool m0 = (a0 >= a1), o0 = (b0 >= b1);
        const bool maskc = (c == 0) ? m0 : !m0;
        const bool diff  = (m0 != o0);
        const int sown = b*2 + c;            // EDT to TP of c   -> fn weighting
        const int soth = b*2 + (1 - c);      // EDT to TP of 1-c == TN of c -> fp
        const float dn = sqrtf(d2[sown*SLICE_ELEMS + h*WW + w]) /
                         sqrtf(__uint_as_float(meta[sown]));
        const float dp = sqrtf(d2[soth*SLICE_ELEMS + h*WW + w]) /
                         sqrtf(__uint_as_float(meta[soth]));
        const float fn_alpha = 1.0f - dn;
        const float fp_alpha = 1.0f - dp;
        const float pc = (c == 0) ? a0 : a1;
        const float lc = (c == 0) ? b0 : b1;
        const float logp = fmaxf(log_sigmoid_f(pc),  -100.0f);
        const float logq = fmaxf(log_sigmoid_f(-pc), -100.0f);
        const float ce = -(lc*logp + (1.0f - lc)*logq);
        const float alpha = diff ? (maskc ? fp_alpha : fn_alpha) : 0.0f;
        vals[r] = alpha * ce;
        cnt += diff ? 1u : 0u;
    }
    // Wave-wide sum via V_WMMA_F32_16X16X4_F32 with B = ones:
    // D[m][n] = rowsum_m (all columns equal); two half-sums combine via xor-16.
    v2f a;  a[0] = vals[0]; a[1] = vals[1];
    v2f bo; bo[0] = 1.0f;   bo[1] = 1.0f;
    v8f cm = {};
    cm = __builtin_amdgcn_wmma_f32_16x16x4_f32(false, a, false, bo,
                                               (short)0, cm, false, false);
    float lane = cm[0]+cm[1]+cm[2]+cm[3]+cm[4]+cm[5]+cm[6]+cm[7];
    const float wavesum = lane + __shfl_xor(lane, 16, 32);

    if ((threadIdx.x & 31) == 0) swave[threadIdx.x >> 5] = wavesum;
    scnt[threadIdx.x] = cnt;
    __syncthreads();
    for (int off = 128; off > 0; off >>= 1) {
        if (threadIdx.x < off) scnt[threadIdx.x] += scnt[threadIdx.x + off];
        __syncthreads();
    }
    if (threadIdx.x == 0) {
        float sblk = 0.0f;
        for (int k = 0; k < 8; ++k) sblk += swave[k];   // fixed order: deterministic
        psum[blockIdx.x] = sblk;
        pcnt[blockIdx.x] = scnt[0];
    }
}

// ------------- deterministic final reduce + divide ---------------------
__global__ void finalize_kernel(const float* __restrict__ psum,
                                const unsigned* __restrict__ pcnt,
                                float* __restrict__ out) {
    __shared__ float ss[256];
    __shared__ unsigned sc[256];
    const int t = threadIdx.x;
    float v = psum[t] + psum[t+256] + psum[t+512] + psum[t+768];
    unsigned n = pcnt[t] + pcnt[t+256] + pcnt[t+512] + pcnt[t+768];
    ss[t] = v; sc[t] = n;
    __syncthreads();
    for (int off = 128; off > 0; off >>= 1) {
        if (t < off) { ss[t] += ss[t+off]; sc[t] += sc[t+off]; }
        __syncthreads();
    }
    if (t == 0) out[0] = ss[0] / (float)sc[0];
}

extern "C" void kernel_launch(void* const* d_in, const int* in_sizes, int n_in,
                              void* d_out, int out_size, void* d_ws, size_t ws_size,
                              hipStream_t stream) {
    const float* pred  = (const float*)d_in[0];
    const float* label = (const float*)d_in[1];
    float* out = (float*)d_out;

    unsigned* meta = (unsigned*)d_ws;                                    // 8 slice maxima
    float* g2      = (float*)((char*)d_ws + 256);                        // 2 MB: g^2 -> D^2
    float* psum    = (float*)((char*)d_ws + 256 + (size_t)NSLICE*SLICE_ELEMS*sizeof(float));
    unsigned* pcnt = (unsigned*)((char*)psum + 1024*sizeof(float));

    hipLaunchKernelGGL(init_kernel,     dim3(1),    dim3(32),  0, stream, meta);
    hipLaunchKernelGGL(edt_cols_kernel, dim3(8),    dim3(256), 0, stream, pred, label, g2);
    hipLaunchKernelGGL(edt_rows_kernel, dim3(2048), dim3(256), 0, stream, g2, meta);
    hipLaunchKernelGGL(loss_kernel,     dim3(1024), dim3(256), 0, stream,
                       pred, label, g2, meta, psum, pcnt);
    hipLaunchKernelGGL(finalize_kernel, dim3(1),    dim3(256), 0, stream, psum, pcnt, out);
}